// LNDS_44968307589567
// MI455X (gfx1250) — compile-verified
//
#include <hip/hip_runtime.h>
#include <math.h>

// ---- problem constants (match reference setup_inputs) ----
#define TSTEPS 2048
#define NB     256
#define NI     64
#define NS     128
#define NO     128
#define DTC    0.1f
#define LDSP   132   // padded LDS row stride (floats) to spread banks
#define ROWT   8     // row tiles per block in the parallel GEMM phases

typedef __attribute__((ext_vector_type(2))) float v2f;
typedef __attribute__((ext_vector_type(8))) float v8f;

static __device__ __forceinline__ v8f wmma4(v2f a, v2f b, v8f c) {
  // V_WMMA_F32_16X16X4_F32 : D(16x16 f32) = A(16x4 f32) * B(4x16 f32) + C
  return __builtin_amdgcn_wmma_f32_16x16x4_f32(
      /*neg_a=*/false, a, /*neg_b=*/false, b,
      /*c_mod=*/(short)0, c, /*reuse_a=*/false, /*reuse_b=*/false);
}

// =====================================================================
// Phase 1: U[t,b,:] = DT * (inp[t,b,:] @ Bm)   written into xt slot
//   rows R = T*B = 524288, K = NI = 64, N = NS = 128
//   block = 256 threads (8 waves); each wave -> ROWT 16x16 output tiles
// =====================================================================
__global__ __launch_bounds__(256) void LNDS_phase1(const float* __restrict__ inp,
                                                   const float* __restrict__ Bm,
                                                   float* __restrict__ U) {
  const int lane = threadIdx.x & 31;
  const int wave = threadIdx.x >> 5;
  const int h = lane >> 4;      // lane half
  const int n = lane & 15;      // N index within tile / A row index
  const int n0 = wave * 16;

  // Preload B operands once per block: Bm[4s+2h + v][n0+n], v=0..1 per slice s
  v2f Bk[16];
#pragma unroll
  for (int s = 0; s < 16; ++s) {
    const int k0 = 4 * s + 2 * h;
    Bk[s].x = Bm[(size_t)(k0 + 0) * NS + n0 + n];
    Bk[s].y = Bm[(size_t)(k0 + 1) * NS + n0 + n];
  }

  const size_t tile0 = (size_t)blockIdx.x * ROWT;
  for (int r = 0; r < ROWT; ++r) {
    const size_t row0 = (tile0 + r) * 16;
    const float* arow = inp + row0 * NI;

    v8f acc0 = {}, acc1 = {};
#pragma unroll
    for (int ss = 0; ss < 8; ++ss) {
      const int s0 = 2 * ss, s1 = 2 * ss + 1;
      v2f a0 = *(const v2f*)&arow[(size_t)n * NI + 4 * s0 + 2 * h];
      acc0 = wmma4(a0, Bk[s0], acc0);
      v2f a1 = *(const v2f*)&arow[(size_t)n * NI + 4 * s1 + 2 * h];
      acc1 = wmma4(a1, Bk[s1], acc1);
    }
    acc0 += acc1;

    float* urow = U + row0 * NS;
#pragma unroll
    for (int g = 0; g < 8; ++g) {
      const int m = g + 8 * h;  // C/D layout: VGPR g holds M = g (+8 upper half)
      urow[(size_t)m * NS + n0 + n] = DTC * acc0[g];
    }
  }
}

// =====================================================================
// Phase 2 (serial scan): x_t = x_{t-1} @ M + U_t,  M = I + DT*A
//   xt buffer already holds U_t; overwritten in place with x_t.
//   16 blocks (one per 16-row batch tile), 8 waves (one per 16-col tile).
//   x_{t-1} double-buffered in LDS; M column slab in registers.
//   U_t software-pipelined 2 steps ahead so the serial WMMA chain never
//   waits on a global load.
// =====================================================================
__global__ __launch_bounds__(256) void LNDS_scan(const float* __restrict__ A,
                                                 const float* __restrict__ x0,
                                                 float* __restrict__ xt) {
  __shared__ float xbuf[2][16 * LDSP];
  const int lane = threadIdx.x & 31;
  const int wave = threadIdx.x >> 5;
  const int h = lane >> 4;
  const int n = lane & 15;
  const int b0 = blockIdx.x * 16;
  const int n0 = wave * 16;

  // M[:, n0..n0+15] as 32 B-operand slices (64 VGPRs)
  v2f Bk[32];
#pragma unroll
  for (int s = 0; s < 32; ++s) {
    const int k0 = 4 * s + 2 * h;
    const int col = n0 + n;
    Bk[s].x = DTC * A[(size_t)(k0 + 0) * NS + col] + ((k0 + 0) == col ? 1.0f : 0.0f);
    Bk[s].y = DTC * A[(size_t)(k0 + 1) * NS + col] + ((k0 + 1) == col ? 1.0f : 0.0f);
  }

  // Seed LDS buffer 0 with x0 tile (each wave writes its 16x16 piece)
#pragma unroll
  for (int g = 0; g < 8; ++g) {
    const int m = g + 8 * h;
    xbuf[0][m * LDSP + n0 + n] = x0[(size_t)(b0 + m) * NS + n0 + n];
  }
  __syncthreads();

  // Prefetch U_0 and U_1 into rotating register sets
  v8f u0, u1, u2;
  {
    const float* g0 = xt + ((size_t)0 * NB + b0) * NS;
    const float* g1 = xt + ((size_t)1 * NB + b0) * NS;
#pragma unroll
    for (int g = 0; g < 8; ++g) {
      const int m = g + 8 * h;
      u0[g] = g0[(size_t)m * NS + n0 + n];
      u1[g] = g1[(size_t)m * NS + n0 + n];
    }
  }

  int cur = 0;
  for (int t = 0; t < TSTEPS; ++t) {
    // issue prefetch of U_{t+2} first; it overlaps the whole WMMA chain
    const int tp = (t + 2 < TSTEPS) ? (t + 2) : (TSTEPS - 1);
    const float* ug = xt + ((size_t)tp * NB + b0) * NS;
#pragma unroll
    for (int g = 0; g < 8; ++g) {
      const int m = g + 8 * h;
      u2[g] = ug[(size_t)m * NS + n0 + n];
    }

    const float* xl = &xbuf[cur][0];
    v8f acc0 = {}, acc1 = {}, acc2 = {}, acc3 = {};
    // 32 K=4 slices split over 4 independent accumulator chains
#pragma unroll
    for (int ss = 0; ss < 8; ++ss) {
      const int s0 = 4 * ss;
      v2f a0 = *(const v2f*)&xl[n * LDSP + 4 * (s0 + 0) + 2 * h];
      acc0 = wmma4(a0, Bk[s0 + 0], acc0);
      v2f a1 = *(const v2f*)&xl[n * LDSP + 4 * (s0 + 1) + 2 * h];
      acc1 = wmma4(a1, Bk[s0 + 1], acc1);
      v2f a2 = *(const v2f*)&xl[n * LDSP + 4 * (s0 + 2) + 2 * h];
      acc2 = wmma4(a2, Bk[s0 + 2], acc2);
      v2f a3 = *(const v2f*)&xl[n * LDSP + 4 * (s0 + 3) + 2 * h];
      acc3 = wmma4(a3, Bk[s0 + 3], acc3);
    }
    acc0 += acc1;
    acc2 += acc3;
    acc0 += acc2;
    acc0 += u0;   // add U_t (prefetched two steps ago)

    float* xg = xt + ((size_t)t * NB + b0) * NS;
    const int nxt = cur ^ 1;
#pragma unroll
    for (int g = 0; g < 8; ++g) {
      const int m = g + 8 * h;
      const float v = acc0[g];
      xg[(size_t)m * NS + n0 + n] = v;   // emit x_t (overwrites U_t)
      xbuf[nxt][m * LDSP + n0 + n] = v;  // feed next step
    }
    __syncthreads();
    cur = nxt;
    u0 = u1;
    u1 = u2;
  }
}

// =====================================================================
// Phase 3: y[t,b,:] = tanh(xt[t,b,:] @ C + d)   (bandwidth-bound)
// =====================================================================
__global__ __launch_bounds__(256) void LNDS_phase3(const float* __restrict__ xt,
                                                   const float* __restrict__ C,
                                                   const float* __restrict__ d,
                                                   float* __restrict__ yt) {
  const int lane = threadIdx.x & 31;
  const int wave = threadIdx.x >> 5;
  const int h = lane >> 4;
  const int n = lane & 15;
  const int n0 = wave * 16;

  // Preload C column slab once per block (64 VGPRs)
  v2f Ck[32];
#pragma unroll
  for (int s = 0; s < 32; ++s) {
    const int k0 = 4 * s + 2 * h;
    Ck[s].x = C[(size_t)(k0 + 0) * NO + n0 + n];
    Ck[s].y = C[(size_t)(k0 + 1) * NO + n0 + n];
  }
  const float dv = d[n0 + n];

  const size_t tile0 = (size_t)blockIdx.x * ROWT;
  for (int r = 0; r < ROWT; ++r) {
    const size_t row0 = (tile0 + r) * 16;
    const float* arow = xt + row0 * NS;

    v8f acc0 = {}, acc1 = {}, acc2 = {}, acc3 = {};
#pragma unroll
    for (int g = 0; g < 8; ++g) acc0[g] = dv;  // bias folded into chain 0

#pragma unroll
    for (int ss = 0; ss < 8; ++ss) {
      const int s0 = 4 * ss;
      v2f a0 = *(const v2f*)&arow[(size_t)n * NS + 4 * (s0 + 0) + 2 * h];
      acc0 = wmma4(a0, Ck[s0 + 0], acc0);
      v2f a1 = *(const v2f*)&arow[(size_t)n * NS + 4 * (s0 + 1) + 2 * h];
      acc1 = wmma4(a1, Ck[s0 + 1], acc1);
      v2f a2 = *(const v2f*)&arow[(size_t)n * NS + 4 * (s0 + 2) + 2 * h];
      acc2 = wmma4(a2, Ck[s0 + 2], acc2);
      v2f a3 = *(const v2f*)&arow[(size_t)n * NS + 4 * (s0 + 3) + 2 * h];
      acc3 = wmma4(a3, Ck[s0 + 3], acc3);
    }
    acc0 += acc1;
    acc2 += acc3;
    acc0 += acc2;

    float* yrow = yt + row0 * NO;
#pragma unroll
    for (int g = 0; g < 8; ++g) {
      const int m = g + 8 * h;
      yrow[(size_t)m * NO + n0 + n] = tanhf(acc0[g]);
    }
  }
}

extern "C" void kernel_launch(void* const* d_in, const int* in_sizes, int n_in,
                              void* d_out, int out_size, void* d_ws, size_t ws_size,
                              hipStream_t stream) {
  (void)in_sizes; (void)n_in; (void)out_size; (void)d_ws; (void)ws_size;
  const float* inp = (const float*)d_in[0];  // [T,B,NI]
  const float* x0  = (const float*)d_in[1];  // [B,NS]
  const float* A   = (const float*)d_in[2];  // [NS,NS]
  const float* Bm  = (const float*)d_in[3];  // [NI,NS]
  const float* C   = (const float*)d_in[4];  // [NS,NO]
  const float* dd  = (const float*)d_in[5];  // [1,NO]

  float* xt = (float*)d_out;                  // [T,B,NS]
  float* yt = xt + (size_t)TSTEPS * NB * NS;  // [T,B,NO]

  const int blocks = (TSTEPS * NB) / (16 * ROWT);  // 4096

  // Phase 1: U = DT * inp @ Bm  -> written into xt slot
  LNDS_phase1<<<blocks, 256, 0, stream>>>(inp, Bm, xt);
  // Phase 2: serial scan, x_t overwrites U_t in place
  LNDS_scan<<<NB / 16, 256, 0, stream>>>(A, x0, xt);
  // Phase 3: y = tanh(xt @ C + d)
  LNDS_phase3<<<blocks, 256, 0, stream>>>(xt, C, dd, yt);
}